// CRCVA_59622736003365
// MI455X (gfx1250) — compile-verified
//
#include <hip/hip_runtime.h>
#include <math.h>

#define V_ 4
#define N_ 2048
#define D_ 128
#define K_ 10

typedef float v2f __attribute__((ext_vector_type(2)));
typedef float v4f __attribute__((ext_vector_type(4)));
typedef float v8f __attribute__((ext_vector_type(8)));

#if __has_builtin(__builtin_amdgcn_global_load_async_to_lds_b128)
#define ROW_ASYNC 1
// Exact param-0 type per hipcc diagnostic: vector_size(16) int, addrspace(1), non-const.
typedef int v4i __attribute__((vector_size(4 * sizeof(int))));
typedef __attribute__((address_space(1))) v4i* async_gptr;
typedef __attribute__((address_space(3))) v4i* async_lptr;
#else
#define ROW_ASYNC 0
#endif

// ---------------------------------------------------------------------------
// Kernel 1: fused Q/K/V projections with full-precision fp32 WMMA (16x16x4).
// out[v] = aligned[v] @ W[v]^T, W row-major (out,in).  B[k,e] = W[e*D + k].
// One wave computes one 16x16 output tile; K loop steps by 4.
// ---------------------------------------------------------------------------
__global__ __launch_bounds__(256) void proj_wmma(
    const float* __restrict__ aligned,
    const float* __restrict__ WQ,
    const float* __restrict__ WK,
    const float* __restrict__ WV,
    float* __restrict__ Qn, float* __restrict__ Kn, float* __restrict__ Vn)
{
    const int TILES_E = D_ / 16;   // 8
    const int TILES_M = N_ / 16;   // 128

    int wid = blockIdx.x * (blockDim.x >> 5) + (threadIdx.x >> 5);
    int te  = wid % TILES_E;
    int tm  = (wid / TILES_E) % TILES_M;
    int pv  = wid / (TILES_E * TILES_M);   // 0..11
    int v   = pv & 3;
    int proj = pv >> 2;                    // 0=Q 1=K 2=V

    const float* W   = (proj == 0) ? WQ : (proj == 1) ? WK : WV;
    float*       out = (proj == 0) ? Qn : (proj == 1) ? Kn : Vn;
    const float* A   = aligned + (size_t)v * N_ * D_;
    W   += (size_t)v * D_ * D_;
    out += (size_t)v * N_ * D_;

    int lane = threadIdx.x & 31;
    int half = lane >> 4;      // lanes 16-31 hold K+2,K+3
    int lm   = lane & 15;
    int m0 = tm * 16, e0 = te * 16;

    const float* arow = A + (size_t)(m0 + lm) * D_ + 2 * half;   // A[m, k..k+1]
    const float* brow = W + (size_t)(e0 + lm) * D_ + 2 * half;   // B[k, e] = W[e, k]

    v8f c = {};
    #pragma unroll 4
    for (int k0 = 0; k0 < D_; k0 += 4) {
        v2f a = *(const v2f*)(arow + k0);
        v2f b = *(const v2f*)(brow + k0);
        c = __builtin_amdgcn_wmma_f32_16x16x4_f32(
                /*neg_a=*/false, a, /*neg_b=*/false, b,
                /*c_mod=*/(short)0, c, /*reuse_a=*/false, /*reuse_b=*/false);
    }

    // C/D layout: VGPR j <-> M = j (lanes 0-15) or M = 8+j (lanes 16-31); N = lm
    float* obase = out + (size_t)(m0 + 8 * half) * D_ + e0 + lm;
    #pragma unroll
    for (int j = 0; j < 8; ++j)
        obase[(size_t)j * D_] = c[j];
}

// ---------------------------------------------------------------------------
// Kernel 2: per-view column mean of Vn (diagonal softmax is uniform 1/N).
// ---------------------------------------------------------------------------
__global__ __launch_bounds__(128) void col_mean(
    const float* __restrict__ Vn, float* __restrict__ meanV)
{
    int p = blockIdx.x, t = threadIdx.x;
    const float* base = Vn + (size_t)p * N_ * D_ + t;
    float s = 0.f;
    for (int m = 0; m < N_; ++m) s += base[(size_t)m * D_];
    meanV[p * D_ + t] = s * (1.0f / N_);
}

// ---------------------------------------------------------------------------
// Kernel 3: one block (128 threads) per (p, n).  For each off-diagonal q:
// stage C row in LDS (async-to-LDS if available, else NT b128 loads),
// extract top-10 by iterative block argmax, compute 10 gathered QK dots,
// softmax-10, sparse-accumulate V.  Memory-bound on C (~192 MB streamed).
// ---------------------------------------------------------------------------
__global__ __launch_bounds__(128) void attn_topk(
    const float* __restrict__ C,
    const float* __restrict__ Qn,
    const float* __restrict__ Kn,
    const float* __restrict__ Vn,
    const float* __restrict__ meanV,
    float* __restrict__ out)
{
    __shared__ float s_row[N_];
    __shared__ float s_q[D_];
    __shared__ float s_val[128];
    __shared__ int   s_idx[128];
    __shared__ int   s_top[K_];
    __shared__ float s_sc[K_];

    const int t = threadIdx.x;
    const int p = blockIdx.x >> 11;        // /N_
    const int n = blockIdx.x & (N_ - 1);
    const float RSQ = 0.08838834764831845f;   // 1/sqrt(128)

    s_q[t] = Qn[((size_t)p * N_ + n) * D_ + t];
    float acc = meanV[p * D_ + t];           // diagonal block contribution

    for (int q = 0; q < V_; ++q) {
        if (q == p) continue;
        const float* crow = C + (((size_t)(p * V_ + q) * N_ + n) * N_);

        __syncthreads();                     // protect s_row/s_top from prev iter
#if ROW_ASYNC
        {
            // CDNA5 async memory->LDS path (ASYNCcnt), 16B per lane per issue.
            #pragma unroll
            for (int i = 0; i < N_ / (128 * 4); ++i) {
                int o = (t + i * 128) * 4;
                __builtin_amdgcn_global_load_async_to_lds_b128(
                    (async_gptr)(crow + o),
                    (async_lptr)(&s_row[o]),
                    0, 0);
            }
#if __has_builtin(__builtin_amdgcn_s_wait_asynccnt)
            __builtin_amdgcn_s_wait_asynccnt(0);
#else
            asm volatile("s_wait_asynccnt 0x0" ::: "memory");
#endif
        }
#else
        {
            const v4f* crow4 = (const v4f*)crow;
            #pragma unroll
            for (int i = 0; i < N_ / (128 * 4); ++i) {
                v4f vv = __builtin_nontemporal_load(crow4 + t + i * 128);
                *(v4f*)&s_row[(t + i * 128) * 4] = vv;
            }
        }
#endif
        __syncthreads();

        // --- top-10 by iterative argmax extraction (C values are in [0,1)) ---
        for (int j = 0; j < K_; ++j) {
            float bv = -1.0f; int bi = 0;
            #pragma unroll
            for (int i = 0; i < N_ / 128; ++i) {
                int m = t + i * 128;
                float vv = s_row[m];
                if (vv > bv) { bv = vv; bi = m; }
            }
            s_val[t] = bv; s_idx[t] = bi;
            __syncthreads();
            #pragma unroll
            for (int off = 64; off > 0; off >>= 1) {
                if (t < off && s_val[t + off] > s_val[t]) {
                    s_val[t] = s_val[t + off]; s_idx[t] = s_idx[t + off];
                }
                __syncthreads();
            }
            if (t == 0) { s_top[j] = s_idx[0]; s_row[s_idx[0]] = -2.0f; }
            __syncthreads();
        }

        // --- 10 gathered dot products, one j per wave round-robin ---
        int wave = t >> 5, lane = t & 31;
        for (int j = wave; j < K_; j += 4) {
            const float* krow = Kn + ((size_t)q * N_ + s_top[j]) * D_;
            float part = 0.f;
            #pragma unroll
            for (int d = 0; d < D_; d += 32)
                part = fmaf(s_q[d + lane], krow[d + lane], part);
            #pragma unroll
            for (int off = 16; off > 0; off >>= 1)
                part += __shfl_xor(part, off, 32);
            if (lane == 0) s_sc[j] = part * RSQ;
        }
        __syncthreads();

        // --- softmax over the 10 surviving scores (exp(NEG-max)==0 in f32) ---
        float mx = s_sc[0];
        #pragma unroll
        for (int j = 1; j < K_; ++j) mx = fmaxf(mx, s_sc[j]);
        float e[K_], sum = 0.f;
        #pragma unroll
        for (int j = 0; j < K_; ++j) { e[j] = expf(s_sc[j] - mx); sum += e[j]; }
        float inv = 1.0f / sum;

        // --- sparse accumulate of V rows ---
        #pragma unroll
        for (int j = 0; j < K_; ++j)
            acc = fmaf(e[j] * inv, Vn[((size_t)q * N_ + s_top[j]) * D_ + t], acc);
    }

    out[((size_t)p * N_ + n) * D_ + t] = acc;
}

// ---------------------------------------------------------------------------
extern "C" void kernel_launch(void* const* d_in, const int* in_sizes, int n_in,
                              void* d_out, int out_size, void* d_ws, size_t ws_size,
                              hipStream_t stream) {
    const float* aligned = (const float*)d_in[0];
    const float* C       = (const float*)d_in[1];
    const float* WQ      = (const float*)d_in[2];
    const float* WK      = (const float*)d_in[3];
    const float* WV      = (const float*)d_in[4];

    const size_t VND = (size_t)V_ * N_ * D_;
    float* Qn    = (float*)d_ws;
    float* Kn    = Qn + VND;
    float* Vn    = Kn + VND;
    float* meanV = Vn + VND;   // V_*D_ floats

    const int totalWaves = 3 * V_ * (N_ / 16) * (D_ / 16);   // 12288
    proj_wmma<<<dim3(totalWaves / 8), dim3(256), 0, stream>>>(
        aligned, WQ, WK, WV, Qn, Kn, Vn);
    col_mean<<<dim3(V_), dim3(128), 0, stream>>>(Vn, meanV);
    attn_topk<<<dim3(V_ * N_), dim3(128), 0, stream>>>(
        C, Qn, Kn, Vn, meanV, (float*)d_out);
}